// FeatureConverge_45251775430767
// MI455X (gfx1250) — compile-verified
//
#include <hip/hip_runtime.h>

typedef __attribute__((ext_vector_type(2))) float v2f;
typedef __attribute__((ext_vector_type(8))) float v8f;

#define B_TOT 32768
#define L_LEN 2048
#define NGRP  4
#define GLEN  64
#define WP    65   // LDS row pitch (floats) to avoid bank conflicts

// ---------------------------------------------------------------------------
// Kernel A: x_site = tanh(tanh(conv1d_3tap(x, wn(v1,g1)) + b1))
// Pure streaming: float4 in / float4 out. Bandwidth-bound.
// ---------------------------------------------------------------------------
__global__ __launch_bounds__(256) void site_kernel(
    const float* __restrict__ x, const float* __restrict__ v1,
    const float* __restrict__ g1, const float* __restrict__ b1,
    float* __restrict__ out_site)
{
  // weight-norm of the 3-tap filter (tiny, broadcast-cached loads)
  float a0 = v1[0], a1 = v1[1], a2 = v1[2];
  float inv = g1[0] * rsqrtf(a0 * a0 + a1 * a1 + a2 * a2);
  float w0 = a0 * inv, w1 = a1 * inv, w2 = a2 * inv;
  float bias = b1[0];

  long long gid = (long long)blockIdx.x * blockDim.x + threadIdx.x; // over B*L/4
  int t4 = (int)(gid & 511);          // L/4 = 512 float4 per row
  long long row = gid >> 9;
  long long base = row * L_LEN + (long long)t4 * 4;

  const float4 xc = *(const float4*)(x + base);
  float xl = (t4 > 0)   ? x[base - 1] : 0.0f;  // zero 'same' padding per row
  float xr = (t4 < 511) ? x[base + 4] : 0.0f;

  float4 o;
  o.x = tanhf(tanhf(w0 * xl   + w1 * xc.x + w2 * xc.y + bias));
  o.y = tanhf(tanhf(w0 * xc.x + w1 * xc.y + w2 * xc.z + bias));
  o.z = tanhf(tanhf(w0 * xc.y + w1 * xc.z + w2 * xc.w + bias));
  o.w = tanhf(tanhf(w0 * xc.z + w1 * xc.w + w2 * xr   + bias));
  *(float4*)(out_site + base) = o;
}

// ---------------------------------------------------------------------------
// Kernel B: gather -> GEMM(W1)+GEMM(W2) -> relu+sum -> GEMM(Wa) -> sigmoid
//           -> mask -> 4->4 3-tap conv (wn(v2,g2)) -> double tanh -> x_bag
// 256 threads = 8 waves; each wave owns 16 xm rows (= 4 batch samples).
// GEMMs via V_WMMA_F32_16X16X4_F32, K=64 accumulated in 16 steps.
// ---------------------------------------------------------------------------
__global__ __launch_bounds__(256) void bag_kernel(
    const float* __restrict__ x, const int* __restrict__ index_all,
    const float* __restrict__ v2, const float* __restrict__ g2,
    const float* __restrict__ b2,
    const float* __restrict__ W1, const float* __restrict__ bl1,
    const float* __restrict__ W2, const float* __restrict__ bl2,
    const float* __restrict__ Wa, const float* __restrict__ ba,
    float* __restrict__ out_bag)
{
  extern __shared__ float lds[];
  float* lW1  = lds;                 // 64 rows x pitch 65
  float* lW2  = lds + 4160;
  float* lWa  = lds + 8320;
  int*   lidx = (int*)(lds + 12480); // 256 gather indices
  float* lw2n = lds + 12736;         // weight-normed v2: 4x4x3
  float* wbase = lds + 12784;        // per-wave tiles: xm(16xWP) + s(16xWP)

  const int tid = threadIdx.x;

  // Stage weights (row-major W[n][k], pitch 65 -> conflict-free column reads)
  for (int e = tid; e < 4096; e += 256) {
    int n = e >> 6, k = e & 63;
    lW1[n * WP + k] = W1[e];
    lW2[n * WP + k] = W2[e];
    lWa[n * WP + k] = Wa[e];
  }
  lidx[tid] = index_all[tid];
  if (tid < 48) { // weight-norm of v2 per output channel
    int o = tid / 12;
    float s = 0.0f;
    for (int e = 0; e < 12; ++e) { float v = v2[o * 12 + e]; s += v * v; }
    lw2n[tid] = g2[o] * v2[tid] * rsqrtf(s);
  }
  __syncthreads();

  const int lane = tid & 31;
  const int wv   = tid >> 5;
  const int mrow = lane & 15;   // M row served by this lane (A fragment)
  const int hi   = lane >> 4;   // half-wave: selects K pair / M+8 in C/D
  float* xm = wbase + wv * (2 * 16 * WP);
  float* sb = xm + 16 * WP;

  const int b0 = (blockIdx.x * 8 + wv) * 4;  // first batch sample of this tile

  // Gather xm tile: 16 rows (sample*4+group) x 64 cols
  for (int e = 0; e < 32; ++e) {
    int elem = e * 32 + lane;
    int m = elem >> 6, j = elem & 63;
    int g = m & 3, bb = b0 + (m >> 2);
    xm[m * WP + j] = x[(long long)bb * L_LEN + lidx[g * 64 + j]];
  }

  const int nbase = lane & 15;

  // GEMM1 + GEMM2 fused: s = relu(xm@W1^T + bl1) + relu(xm@W2^T + bl2)
  for (int nt = 0; nt < 4; ++nt) {
    int n = nbase + 16 * nt;
    float c1 = bl1[n], c2 = bl2[n];
    v8f acc1, acc2;
    for (int r = 0; r < 8; ++r) { acc1[r] = c1; acc2[r] = c2; }
    for (int kk = 0; kk < 16; ++kk) {
      int c = 4 * kk + 2 * hi;
      v2f a;  a.x  = xm[mrow * WP + c];  a.y  = xm[mrow * WP + c + 1];
      v2f bA; bA.x = lW1[n * WP + c];    bA.y = lW1[n * WP + c + 1];
      v2f bB; bB.x = lW2[n * WP + c];    bB.y = lW2[n * WP + c + 1];
      acc1 = __builtin_amdgcn_wmma_f32_16x16x4_f32(false, a, false, bA,
                                                   (short)0, acc1, false, false);
      acc2 = __builtin_amdgcn_wmma_f32_16x16x4_f32(false, a, false, bB,
                                                   (short)0, acc2, false, false);
    }
    for (int r = 0; r < 8; ++r) {
      float s = fmaxf(acc1[r], 0.0f) + fmaxf(acc2[r], 0.0f);
      sb[(r + 8 * hi) * WP + n] = s;   // D-layout -> row-major LDS
    }
  }

  // GEMM3: A = sigmoid(s@Wa^T + ba); xmask = (A+1)*xm (overwrite xm in place)
  for (int nt = 0; nt < 4; ++nt) {
    int n = nbase + 16 * nt;
    float cb = ba[n];
    v8f acc;
    for (int r = 0; r < 8; ++r) acc[r] = cb;
    for (int kk = 0; kk < 16; ++kk) {
      int c = 4 * kk + 2 * hi;
      v2f a;  a.x  = sb[mrow * WP + c];  a.y  = sb[mrow * WP + c + 1];
      v2f bA; bA.x = lWa[n * WP + c];    bA.y = lWa[n * WP + c + 1];
      acc = __builtin_amdgcn_wmma_f32_16x16x4_f32(false, a, false, bA,
                                                  (short)0, acc, false, false);
    }
    for (int r = 0; r < 8; ++r) {
      int m = r + 8 * hi;
      float att = 1.0f / (1.0f + __expf(-acc[r]));
      xm[m * WP + n] = (att + 1.0f) * xm[m * WP + n];
    }
  }

  // 4->4 channel 3-tap conv over GL (zero pad) + double tanh -> x_bag
  for (int e = 0; e < 32; ++e) {
    int elem = e * 32 + lane;
    int t = elem & 63;
    int m = elem >> 6;
    int o = m & 3, ss = m >> 2;
    float acc = b2[o];
    for (int i = 0; i < 4; ++i) {
      const float* rowp = xm + (ss * 4 + i) * WP;
      float xl = (t > 0)  ? rowp[t - 1] : 0.0f;
      float xc = rowp[t];
      float xr = (t < 63) ? rowp[t + 1] : 0.0f;
      const float* wp = lw2n + (o * 4 + i) * 3;
      acc += wp[0] * xl + wp[1] * xc + wp[2] * xr;
    }
    out_bag[(long long)(b0 + ss) * (NGRP * GLEN) + o * GLEN + t] =
        tanhf(tanhf(acc));
  }
}

// ---------------------------------------------------------------------------
extern "C" void kernel_launch(void* const* d_in, const int* in_sizes, int n_in,
                              void* d_out, int out_size, void* d_ws, size_t ws_size,
                              hipStream_t stream) {
  const float* x         = (const float*)d_in[0];
  const int*   index_all = (const int*)  d_in[1];
  const float* v1        = (const float*)d_in[2];
  const float* g1        = (const float*)d_in[3];
  const float* b1        = (const float*)d_in[4];
  const float* v2        = (const float*)d_in[5];
  const float* g2        = (const float*)d_in[6];
  const float* b2        = (const float*)d_in[7];
  const float* W1        = (const float*)d_in[8];
  const float* bl1       = (const float*)d_in[9];
  const float* W2        = (const float*)d_in[10];
  const float* bl2       = (const float*)d_in[11];
  const float* Wa        = (const float*)d_in[12];
  const float* ba        = (const float*)d_in[13];

  float* out_bag  = (float*)d_out;                              // B*4*64
  float* out_site = out_bag + (size_t)B_TOT * NGRP * GLEN;      // B*2048

  // x_bag pipeline: B/32 workgroups x 256 threads, ~115 KB dynamic LDS
  size_t ldsBytes = (size_t)(12784 + 8 * 2 * 16 * WP) * sizeof(float);
  hipFuncSetAttribute((const void*)bag_kernel,
                      hipFuncAttributeMaxDynamicSharedMemorySize, (int)ldsBytes);
  hipLaunchKernelGGL(bag_kernel, dim3(B_TOT / 32), dim3(256), ldsBytes, stream,
                     x, index_all, v2, g2, b2, W1, bl1, W2, bl2, Wa, ba, out_bag);

  // x_site streaming conv: B*L/4 float4 threads
  long long n4 = (long long)B_TOT * L_LEN / 4;
  hipLaunchKernelGGL(site_kernel, dim3((unsigned)(n4 / 256)), dim3(256), 0, stream,
                     x, v1, g1, b1, out_site);
}